// GaussianStep_14740327759970
// MI455X (gfx1250) — compile-verified
//
#include <hip/hip_runtime.h>

typedef __attribute__((ext_vector_type(2))) float v2f;
typedef __attribute__((ext_vector_type(8))) float v8f;

#define G_TOT 100000
#define WPB 8              // waves per block (wave32)
#define LDSF 1968          // floats per wave slice (16B-aligned regions)

// per-wave LDS slice offsets (floats); all multiples of 4 -> 16B aligned
#define O_COV 0     // 16x16
#define O_HP  256   // H zero-padded to 16x16 (rows 8..15 = 0)
#define O_RP  512   // R zero-padded to 16x16 (rows/cols 8..15 = 0)
#define O_F   768   // 16x16
#define O_Q   1024  // 16x16
#define O_T1  1280  // 16x16 scratch
#define O_T2  1536  // 16x16 scratch: sys first, ImKH later
#define O_K   1792  // 16x8  (K, row-major [s][m])
#define O_VEC 1920  // mean[0..15], obs/innov[16..23], new_mean[24..39]

__device__ __forceinline__ v8f wmma4(v2f a, v2f b, v8f c) {
  // D(16x16,f32) = A(16x4,f32) * B(4x16,f32) + C
  return __builtin_amdgcn_wmma_f32_16x16x4_f32(
      false, a, false, b, (short)0, c, false, false);
}

// Wave-level LDS sync: same-wave DS ops are in-order at the LDS unit (ISA
// ch.11), so only a compiler reordering fence is needed — no s_barrier.
__device__ __forceinline__ void wsync() {
  __builtin_amdgcn_fence(__ATOMIC_SEQ_CST, "wavefront");
  __builtin_amdgcn_wave_barrier();
  __builtin_amdgcn_fence(__ATOMIC_SEQ_CST, "wavefront");
}

// broadcast from a constant lane (folds to v_readlane_b32 -> SGPR operand)
__device__ __forceinline__ float rdl(float x, int l) {
  return __builtin_bit_cast(
      float, __builtin_amdgcn_readlane(__builtin_bit_cast(int, x), l));
}

__global__ void __launch_bounds__(256)
kalman_step_kernel(const float* __restrict__ g_obs,   // [G,8]
                   const float* __restrict__ g_mean,  // [G,16]
                   const float* __restrict__ g_cov,   // [G,16,16]
                   const float* __restrict__ g_H,     // [G,8,16]
                   const float* __restrict__ g_R,     // [G,8,8]
                   const float* __restrict__ g_F,     // [G,16,16]
                   const float* __restrict__ g_Q,     // [G,16,16]
                   float* __restrict__ o_mean,        // [G,16]
                   float* __restrict__ o_cov)         // [G,16,16]
{
  __shared__ __align__(16) float lds[WPB * LDSF];
  const int wave = threadIdx.x >> 5;
  const int lane = threadIdx.x & 31;
  const int g = blockIdx.x * WPB + wave;

  float* w = &lds[wave * LDSF];

  const int lo = lane & 15;            // col for B/C/D frags, row for A frags
  const int hi = (lane >> 4) & 1;      // half-wave select
  const int kbase = hi * 2;            // A/B k offset within a 4-block
  const int crow = hi * 8;             // C/D base row

  // ---------------- stage inputs into per-wave LDS (float4) ----------------
  {
    const float4 z4 = make_float4(0.f, 0.f, 0.f, 0.f);
    const float4* pc = (const float4*)(g_cov + (size_t)g * 256);
    float4* c4 = (float4*)&w[O_COV];
    c4[lane] = pc[lane]; c4[lane + 32] = pc[lane + 32];

    const float4* ph = (const float4*)(g_H + (size_t)g * 128);
    float4* h4 = (float4*)&w[O_HP];
    h4[lane] = ph[lane];          // rows 0..7
    h4[lane + 32] = z4;           // rows 8..15 zero

    // R (8x8) -> 16x16 zero-padded tile
    const float4* pr = (const float4*)(g_R + (size_t)g * 64);  // 16 float4s
    float4* r4 = (float4*)&w[O_RP];
    {
      const int row = lane >> 2;         // 0..7
      const int c = lane & 3;            // f4-column 0..3
      float4 rv = pr[row * 2 + (c & 1)]; // in-range gather
      if (c >= 2) rv = z4;               // cols 8..15 zero
      r4[lane] = rv;                     // rows 0..7
      r4[lane + 32] = z4;                // rows 8..15 zero
    }

    const float4* pf = (const float4*)(g_F + (size_t)g * 256);
    float4* f4 = (float4*)&w[O_F];
    f4[lane] = pf[lane]; f4[lane + 32] = pf[lane + 32];

    const float4* pq = (const float4*)(g_Q + (size_t)g * 256);
    float4* q4 = (float4*)&w[O_Q];
    q4[lane] = pq[lane]; q4[lane + 32] = pq[lane + 32];

    if (lane < 16) w[O_VEC + lane] = g_mean[(size_t)g * 16 + lane];
    if (lane < 8)  w[O_VEC + 16 + lane] = g_obs[(size_t)g * 8 + lane];
  }
  wsync();

  // ---------------- T1 = cov @ H^T  (16x16, cols 8..15 are 0) ----------------
  {
    v8f acc = {};
#pragma unroll
    for (int kb = 0; kb < 4; ++kb) {
      const int k0 = kb * 4 + kbase;
      v2f a; a.x = w[O_COV + lo * 16 + k0]; a.y = w[O_COV + lo * 16 + k0 + 1];
      v2f b;   // B(k,n) = Hpad[n][k]  (rows n>=8 are zero)
      b.x = w[O_HP + lo * 16 + k0]; b.y = w[O_HP + lo * 16 + k0 + 1];
      acc = wmma4(a, b, acc);
    }
#pragma unroll
    for (int v = 0; v < 8; ++v) w[O_T1 + (crow + v) * 16 + lo] = acc[v];
  }
  wsync();

  // ---------------- sys = Hpad @ T1 + Rpad  (valid 8x8 in top-left) -> T2 ----
  {
    v8f sf;
#pragma unroll
    for (int v = 0; v < 8; ++v) sf[v] = w[O_RP + (crow + v) * 16 + lo];
#pragma unroll
    for (int kb = 0; kb < 4; ++kb) {
      const int k0 = kb * 4 + kbase;
      v2f a; a.x = w[O_HP + lo * 16 + k0]; a.y = w[O_HP + lo * 16 + k0 + 1];
      v2f b; b.x = w[O_T1 + k0 * 16 + lo]; b.y = w[O_T1 + (k0 + 1) * 16 + lo];
      sf = wmma4(a, b, sf);
    }
#pragma unroll
    for (int v = 0; v < 8; ++v) w[O_T2 + (crow + v) * 16 + lo] = sf[v];
  }
  wsync();

  // ---- register Cholesky of sys: lane i holds row i; readlane broadcasts ----
  float r[8], dinv[8];
  {
    const int row = lane & 7;
#pragma unroll
    for (int p = 0; p < 8; ++p) r[p] = w[O_T2 + row * 16 + p];
#pragma unroll
    for (int k = 0; k < 8; ++k) {
      float s = r[k];
#pragma unroll
      for (int p = 0; p < k; ++p) s -= r[p] * rdl(r[p], k);
      const float d = rdl(s, k);                       // pivot from lane k
      const float diag = __builtin_amdgcn_sqrtf(fmaxf(d, 1e-20f));
      const float inv = __builtin_amdgcn_rcpf(diag);   // every lane keeps it
      dinv[k] = inv;
      const float nv = (row == k) ? diag : s * inv;
      r[k] = (row >= k) ? nv : r[k];                   // cndmask, no branches
    }
  }

  // ---- K = PHt * sys^-1: lane n solves L L^T z = PHt[n,:], multiplies only --
  {
    const int n = lane & 15;            // lanes 16..31 duplicate (discarded)
    float y[8];
#pragma unroll
    for (int i = 0; i < 8; ++i) {
      float s = w[O_T1 + n * 16 + i];   // PHt[n][i]
#pragma unroll
      for (int p = 0; p < i; ++p) s -= rdl(r[p], i) * y[p];   // L[i][p]
      y[i] = s * dinv[i];
    }
#pragma unroll
    for (int i = 7; i >= 0; --i) {
      float s = y[i];
#pragma unroll
      for (int p = i + 1; p < 8; ++p) s -= rdl(r[i], p) * y[p]; // L[p][i]
      y[i] = s * dinv[i];
    }
    if (lane < 16) {
#pragma unroll
      for (int i = 0; i < 8; ++i) w[O_K + lane * 8 + i] = y[i];  // K[n][i]
    }
  }
  wsync();

  // ---------------- mean update ----------------
  if (lane < 8) {           // innov = obs - H @ mean
    float s = 0.0f;
#pragma unroll
    for (int t = 0; t < 16; ++t) s += w[O_HP + lane * 16 + t] * w[O_VEC + t];
    w[O_VEC + 16 + lane] = w[O_VEC + 16 + lane] - s;
  }
  wsync();
  if (lane < 16) {          // new_mean = mean + K @ innov
    float s = w[O_VEC + lane];
#pragma unroll
    for (int m = 0; m < 8; ++m) s += w[O_K + lane * 8 + m] * w[O_VEC + 16 + m];
    w[O_VEC + 24 + lane] = s;
  }
  wsync();

  // ---------------- ImKH = I - K @ H  -> T2 (overwrites sys) ----------------
  {
    v8f kh = {};
#pragma unroll
    for (int kb = 0; kb < 2; ++kb) {
      const int k0 = kb * 4 + kbase;
      v2f a; a.x = w[O_K + lo * 8 + k0]; a.y = w[O_K + lo * 8 + k0 + 1];
      v2f b; b.x = w[O_HP + k0 * 16 + lo]; b.y = w[O_HP + (k0 + 1) * 16 + lo];
      kh = wmma4(a, b, kh);
    }
#pragma unroll
    for (int v = 0; v < 8; ++v) {
      const int m = crow + v;
      w[O_T2 + m * 16 + lo] = ((m == lo) ? 1.0f : 0.0f) - kh[v];
    }
  }
  wsync();

  // ---------------- T1 = ImKH @ cov ----------------
  {
    v8f acc = {};
#pragma unroll
    for (int kb = 0; kb < 4; ++kb) {
      const int k0 = kb * 4 + kbase;
      v2f a; a.x = w[O_T2 + lo * 16 + k0]; a.y = w[O_T2 + lo * 16 + k0 + 1];
      v2f b; b.x = w[O_COV + k0 * 16 + lo]; b.y = w[O_COV + (k0 + 1) * 16 + lo];
      acc = wmma4(a, b, acc);
    }
#pragma unroll
    for (int v = 0; v < 8; ++v) w[O_T1 + (crow + v) * 16 + lo] = acc[v];
  }
  wsync();

  // ---------------- ncov = T1 @ ImKH^T ----------------
  v8f ncov = {};
#pragma unroll
  for (int kb = 0; kb < 4; ++kb) {
    const int k0 = kb * 4 + kbase;
    v2f a; a.x = w[O_T1 + lo * 16 + k0]; a.y = w[O_T1 + lo * 16 + k0 + 1];
    v2f b;  // (ImKH^T)(k,n) = ImKH[n][k]
    b.x = w[O_T2 + lo * 16 + k0]; b.y = w[O_T2 + lo * 16 + k0 + 1];
    ncov = wmma4(a, b, ncov);
  }
  wsync();

  // ---------------- T1 = Rpad(16x16) @ K^T  (rows 8..15 zero) ---------------
  {
    v8f rkt = {};
#pragma unroll
    for (int kb = 0; kb < 2; ++kb) {
      const int k0 = kb * 4 + kbase;
      v2f a;  // A(m,k) = Rpad[m][k]  (rows m>=8 are zero)
      a.x = w[O_RP + lo * 16 + k0]; a.y = w[O_RP + lo * 16 + k0 + 1];
      v2f b;  // (K^T)(k,n) = K[n][k]
      b.x = w[O_K + lo * 8 + k0]; b.y = w[O_K + lo * 8 + k0 + 1];
      rkt = wmma4(a, b, rkt);
    }
#pragma unroll
    for (int v = 0; v < 8; ++v) w[O_T1 + (crow + v) * 16 + lo] = rkt[v];
  }
  wsync();

  // ---------------- ncov += K @ (R K^T) ----------------
#pragma unroll
  for (int kb = 0; kb < 2; ++kb) {
    const int k0 = kb * 4 + kbase;
    v2f a; a.x = w[O_K + lo * 8 + k0]; a.y = w[O_K + lo * 8 + k0 + 1];
    v2f b; b.x = w[O_T1 + k0 * 16 + lo]; b.y = w[O_T1 + (k0 + 1) * 16 + lo];
    ncov = wmma4(a, b, ncov);
  }
#pragma unroll
  for (int v = 0; v < 8; ++v) w[O_T2 + (crow + v) * 16 + lo] = ncov[v];
  wsync();

  // ---------------- T1 = F @ ncov ----------------
  {
    v8f acc = {};
#pragma unroll
    for (int kb = 0; kb < 4; ++kb) {
      const int k0 = kb * 4 + kbase;
      v2f a; a.x = w[O_F + lo * 16 + k0]; a.y = w[O_F + lo * 16 + k0 + 1];
      v2f b; b.x = w[O_T2 + k0 * 16 + lo]; b.y = w[O_T2 + (k0 + 1) * 16 + lo];
      acc = wmma4(a, b, acc);
    }
#pragma unroll
    for (int v = 0; v < 8; ++v) w[O_T1 + (crow + v) * 16 + lo] = acc[v];
  }
  wsync();

  // ---------------- pred_cov = T1 @ F^T + Q ----------------
  {
    v8f pc;
#pragma unroll
    for (int v = 0; v < 8; ++v) pc[v] = w[O_Q + (crow + v) * 16 + lo];
#pragma unroll
    for (int kb = 0; kb < 4; ++kb) {
      const int k0 = kb * 4 + kbase;
      v2f a; a.x = w[O_T1 + lo * 16 + k0]; a.y = w[O_T1 + lo * 16 + k0 + 1];
      v2f b;  // (F^T)(k,n) = F[n][k]
      b.x = w[O_F + lo * 16 + k0]; b.y = w[O_F + lo * 16 + k0 + 1];
      pc = wmma4(a, b, pc);
    }
    // stage result in LDS, then vectorized b128 stores to global
#pragma unroll
    for (int v = 0; v < 8; ++v) w[O_T2 + (crow + v) * 16 + lo] = pc[v];
  }
  wsync();
  {
    float4* oc4 = (float4*)(o_cov + (size_t)g * 256);
    const float4* t2 = (const float4*)&w[O_T2];
    oc4[lane] = t2[lane];
    oc4[lane + 32] = t2[lane + 32];
  }

  // ---------------- pred_mean = F @ new_mean -> global ----------------
  if (lane < 16) {
    float s = 0.0f;
#pragma unroll
    for (int t = 0; t < 16; ++t) s += w[O_F + lane * 16 + t] * w[O_VEC + 24 + t];
    o_mean[(size_t)g * 16 + lane] = s;
  }
}

extern "C" void kernel_launch(void* const* d_in, const int* in_sizes, int n_in,
                              void* d_out, int out_size, void* d_ws, size_t ws_size,
                              hipStream_t stream) {
  (void)in_sizes; (void)n_in; (void)out_size; (void)d_ws; (void)ws_size;
  const float* obs  = (const float*)d_in[0];  // [G,8]
  const float* mean = (const float*)d_in[1];  // [G,16]
  const float* cov  = (const float*)d_in[2];  // [G,16,16]
  const float* H    = (const float*)d_in[3];  // [G,8,16]
  const float* R    = (const float*)d_in[4];  // [G,8,8]
  const float* F    = (const float*)d_in[5];  // [G,16,16]
  const float* Q    = (const float*)d_in[6];  // [G,16,16]

  float* out_mean = (float*)d_out;                       // [G,16]
  float* out_cov  = (float*)d_out + (size_t)G_TOT * 16;  // [G,16,16]

  const int blocks = G_TOT / WPB;  // 100000 / 8 = 12500, exact
  kalman_step_kernel<<<blocks, WPB * 32, 0, stream>>>(
      obs, mean, cov, H, R, F, Q, out_mean, out_cov);
}